// DiffClusterMIST_66486093742442
// MI455X (gfx1250) — compile-verified
//
#include <hip/hip_runtime.h>
#include <hip/hip_bf16.h>
#include <math.h>

// ---------------------------------------------------------------------------
// KSG mutual information estimator, fused.  N=8192, D=256, K=3, C=10.
// Pass 1: fp32 WMMA GEMM tiles -> same-class top-4 smallest d^2 -> anchor^2
// Pass 2: fp32 WMMA GEMM tiles -> count d^2 < anchor^2 -> digamma(m_i+eps)
// Pass 3: deterministic reduction + class histogram + scalar MI.
// ---------------------------------------------------------------------------

#define NPTS   8192
#define DDIM   256
#define KNN    3
#define NCLS   10
#define NSTRIP (NPTS / 16)   // 512 row strips
#define NJBLK  (NPTS / 64)   // 128 column blocks of 64

typedef float v2f __attribute__((ext_vector_type(2)));
typedef float v8f __attribute__((ext_vector_type(8)));

#if defined(__HIP_DEVICE_COMPILE__) && defined(__gfx1250__) && \
    __has_builtin(__builtin_amdgcn_wmma_f32_16x16x4_f32)
#define USE_WMMA 1
#else
#define USE_WMMA 0
#endif

__device__ __forceinline__ float digammaf_dev(float x) {
  // recurrence to x >= 6, then asymptotic series (matches jax digamma closely)
  float r = 0.0f;
  while (x < 6.0f) { r -= 1.0f / x; x += 1.0f; }
  float inv  = 1.0f / x;
  float inv2 = inv * inv;
  r += __logf(x) - 0.5f * inv
       - inv2 * (1.0f / 12.0f - inv2 * (1.0f / 120.0f - inv2 * (1.0f / 252.0f)));
  return r;
}

// ---------------------------------------------------------------------------
// sq[i] = sum_k X[i][k]^2   (one wave per row, 8 rows per 256-thread block)
// ---------------------------------------------------------------------------
__global__ void sq_kernel(const float* __restrict__ X, float* __restrict__ sq) {
  const int row  = blockIdx.x * 8 + (threadIdx.x >> 5);
  const int lane = threadIdx.x & 31;
  const float4* xr = (const float4*)(X + (size_t)row * DDIM);
  float4 a = xr[lane];
  float4 b = xr[lane + 32];
  float s = a.x * a.x + a.y * a.y + a.z * a.z + a.w * a.w +
            b.x * b.x + b.y * b.y + b.z * b.z + b.w * b.w;
#pragma unroll
  for (int off = 16; off > 0; off >>= 1) s += __shfl_down(s, off, 32);
  if (lane == 0) sq[row] = s;
}

// ---------------------------------------------------------------------------
// Core GEMM block: one wave, 16 rows (strip i0) x 64 cols (block j0).
// acc[q][v] = dot(X[i0 + 8*half + v], X[j0 + 16*q + nloc])  in WMMA layout:
//   lane l: nloc = l&15 (N), half = l>>4; C vgpr v holds M = 8*half + v.
// ---------------------------------------------------------------------------
__device__ __forceinline__ void gemm_block(const float* __restrict__ X,
                                           int i0, int j0, int nloc, int half,
                                           v8f acc[4]) {
  const v8f z = {0.f, 0.f, 0.f, 0.f, 0.f, 0.f, 0.f, 0.f};
#pragma unroll
  for (int q = 0; q < 4; ++q) acc[q] = z;

#if USE_WMMA
  const float* Arow  = X + (size_t)(i0 + nloc) * DDIM + 2 * half;
  const float* Brow0 = X + (size_t)(j0 +  0 + nloc) * DDIM + 2 * half;
  const float* Brow1 = X + (size_t)(j0 + 16 + nloc) * DDIM + 2 * half;
  const float* Brow2 = X + (size_t)(j0 + 32 + nloc) * DDIM + 2 * half;
  const float* Brow3 = X + (size_t)(j0 + 48 + nloc) * DDIM + 2 * half;
#pragma unroll 8
  for (int k0 = 0; k0 < DDIM; k0 += 4) {
    v2f a  = *(const v2f*)(Arow  + k0);
    v2f b0 = *(const v2f*)(Brow0 + k0);
    v2f b1 = *(const v2f*)(Brow1 + k0);
    v2f b2 = *(const v2f*)(Brow2 + k0);
    v2f b3 = *(const v2f*)(Brow3 + k0);
    acc[0] = __builtin_amdgcn_wmma_f32_16x16x4_f32(false, a, false, b0,
                                                   (short)0, acc[0], false, false);
    acc[1] = __builtin_amdgcn_wmma_f32_16x16x4_f32(false, a, false, b1,
                                                   (short)0, acc[1], false, false);
    acc[2] = __builtin_amdgcn_wmma_f32_16x16x4_f32(false, a, false, b2,
                                                   (short)0, acc[2], false, false);
    acc[3] = __builtin_amdgcn_wmma_f32_16x16x4_f32(false, a, false, b3,
                                                   (short)0, acc[3], false, false);
  }
#else
  // Scalar fallback (only if the f32 WMMA builtin is unavailable).
  for (int q = 0; q < 4; ++q) {
    const float* br = X + (size_t)(j0 + 16 * q + nloc) * DDIM;
    for (int v = 0; v < 8; ++v) {
      const float* ar = X + (size_t)(i0 + 8 * half + v) * DDIM;
      float d = 0.f;
      for (int k = 0; k < DDIM; ++k) d += ar[k] * br[k];
      acc[q][v] = d;
    }
  }
#endif
}

// ---------------------------------------------------------------------------
// Pass 1: anchor^2[i] = 4th smallest same-class squared distance (self incl.)
// One wave (32 threads) per 16-row strip.
// ---------------------------------------------------------------------------
__global__ void anchor_kernel(const float* __restrict__ X,
                              const int* __restrict__ y,
                              const float* __restrict__ sq,
                              float* __restrict__ anchor2) {
  const int i0   = blockIdx.x * 16;
  const int lane = threadIdx.x;
  const int nloc = lane & 15;
  const int half = lane >> 4;

  float sqi[8];
  int   yi[8];
#pragma unroll
  for (int v = 0; v < 8; ++v) {
    int r = i0 + 8 * half + v;
    sqi[v] = sq[r];
    yi[v]  = y[r];
  }

  float t0 = __builtin_inff(), t1 = t0, t2 = t0, t3 = t0;  // top-4 (ascending)
  __shared__ float tile[16][68];

  for (int jt = 0; jt < NJBLK; ++jt) {
    const int j0 = jt * 64;
    if (jt + 1 < NJBLK)
      __builtin_prefetch(X + (size_t)(j0 + 64 + nloc) * DDIM + 2 * half, 0, 3);

    v8f acc[4];
    gemm_block(X, i0, j0, nloc, half, acc);

#pragma unroll
    for (int q = 0; q < 4; ++q) {
      float sqj = sq[j0 + 16 * q + nloc];
      int   yj  = y [j0 + 16 * q + nloc];
#pragma unroll
      for (int v = 0; v < 8; ++v) {
        float d2 = sqi[v] + sqj - 2.0f * acc[q][v];
        d2 = d2 > 0.f ? d2 : 0.f;
        if (yi[v] != yj) d2 = __builtin_inff();
        tile[8 * half + v][16 * q + nloc] = d2;
      }
    }
    __syncthreads();
    if (lane < 16) {
#pragma unroll 8
      for (int n = 0; n < 64; ++n) {
        float vv = tile[lane][n];
        if (vv < t3) {
          t3 = vv;
          if (t3 < t2) { float t = t2; t2 = t3; t3 = t; }
          if (t2 < t1) { float t = t1; t1 = t2; t2 = t; }
          if (t1 < t0) { float t = t0; t0 = t1; t1 = t; }
        }
      }
    }
    __syncthreads();
  }
  if (lane < 16) anchor2[i0 + lane] = t3;   // (K+1)-th smallest, K=3
}

// ---------------------------------------------------------------------------
// Pass 2: m_i = #{j : d2[i][j] < anchor2[i]} - 1 ; dig[i] = digamma(m_i+1e-7)
// ---------------------------------------------------------------------------
__global__ void count_kernel(const float* __restrict__ X,
                             const float* __restrict__ sq,
                             const float* __restrict__ anchor2,
                             float* __restrict__ dig) {
  const int i0   = blockIdx.x * 16;
  const int lane = threadIdx.x;
  const int nloc = lane & 15;
  const int half = lane >> 4;

  float sqi[8], anc[8];
#pragma unroll
  for (int v = 0; v < 8; ++v) {
    int r = i0 + 8 * half + v;
    sqi[v] = sq[r];
    anc[v] = anchor2[r];
  }

  float cnt[8] = {0.f, 0.f, 0.f, 0.f, 0.f, 0.f, 0.f, 0.f};

  for (int jt = 0; jt < NJBLK; ++jt) {
    const int j0 = jt * 64;
    if (jt + 1 < NJBLK)
      __builtin_prefetch(X + (size_t)(j0 + 64 + nloc) * DDIM + 2 * half, 0, 3);

    v8f acc[4];
    gemm_block(X, i0, j0, nloc, half, acc);

#pragma unroll
    for (int q = 0; q < 4; ++q) {
      float sqj = sq[j0 + 16 * q + nloc];
#pragma unroll
      for (int v = 0; v < 8; ++v) {
        float d2 = sqi[v] + sqj - 2.0f * acc[q][v];
        d2 = d2 > 0.f ? d2 : 0.f;
        cnt[v] += (d2 < anc[v]) ? 1.0f : 0.0f;
      }
    }
  }

  // transpose counts so lanes 0..15 each own one full row, then digamma
  __shared__ float cbuf[8][32];
#pragma unroll
  for (int v = 0; v < 8; ++v) cbuf[v][lane] = cnt[v];
  __syncthreads();
  if (lane < 16) {
    const int h = lane >> 3, vv = lane & 7;
    float s = 0.f;
#pragma unroll
    for (int n = 0; n < 16; ++n) s += cbuf[vv][h * 16 + n];
    float m = s - 1.0f;                       // exclude self
    dig[i0 + lane] = digammaf_dev(m + 1e-7f);
  }
}

// ---------------------------------------------------------------------------
// Pass 3: deterministic reduction + class histogram + final scalar MI
// ---------------------------------------------------------------------------
__global__ void finalize_kernel(const float* __restrict__ dig,
                                const int* __restrict__ y,
                                float* __restrict__ out) {
  __shared__ float sred[256];
  __shared__ int   scnt[NCLS];
  const int t = threadIdx.x;

  float s = 0.f;
  for (int i = t; i < NPTS; i += 256) s += dig[i];
  sred[t] = s;
  if (t < NCLS) scnt[t] = 0;
  __syncthreads();

  for (int i = t; i < NPTS; i += 256) atomicAdd(&scnt[y[i]], 1);

  for (int off = 128; off > 0; off >>= 1) {
    __syncthreads();
    if (t < off) sred[t] += sred[t + off];
  }
  __syncthreads();

  if (t == 0) {
    const float Nf = (float)NPTS;
    float avg_m = sred[0] / Nf;
    float avg_Nx = 0.f;
    for (int c = 0; c < NCLS; ++c) {
      float nx = (float)scnt[c];
      avg_Nx += (nx / Nf) * digammaf_dev(nx);
    }
    float mi = (digammaf_dev(Nf) - avg_Nx + digammaf_dev((float)KNN) - avg_m)
               / __logf(2.0f);
    out[0] = mi > 0.f ? mi : 0.f;
  }
}

// ---------------------------------------------------------------------------
extern "C" void kernel_launch(void* const* d_in, const int* in_sizes, int n_in,
                              void* d_out, int out_size, void* d_ws, size_t ws_size,
                              hipStream_t stream) {
  const float* X = (const float*)d_in[0];
  const int*   y = (const int*)d_in[1];
  float*       out = (float*)d_out;

  float* sq      = (float*)d_ws;            // 8192 floats
  float* anchor2 = sq + NPTS;               // 8192 floats
  float* dig     = anchor2 + NPTS;          // 8192 floats   (96 KB total)

  sq_kernel     <<<NPTS / 8, 256, 0, stream>>>(X, sq);
  anchor_kernel <<<NSTRIP,    32, 0, stream>>>(X, y, sq, anchor2);
  count_kernel  <<<NSTRIP,    32, 0, stream>>>(X, sq, anchor2, dig);
  finalize_kernel<<<1,       256, 0, stream>>>(dig, y, out);
}